// InteractionBlock_41566693490846
// MI455X (gfx1250) — compile-verified
//
#include <hip/hip_runtime.h>

// ---------------------------------------------------------------------------
// e3nn InteractionBlock for MI455X (gfx1250, wave32, WMMA).
// fp32 throughout via V_WMMA_F32_16X16X4_F32 (exact fp32; problem is
// HBM-bound at ~0.5 GB traffic, so the fp32 matrix path is already fast).
// Async global->LDS tile fills via GLOBAL_LOAD_ASYNC_TO_LDS_B128 (ASYNCcnt).
// ---------------------------------------------------------------------------

typedef float v2f __attribute__((ext_vector_type(2)));
typedef float v8f __attribute__((ext_vector_type(8)));
typedef int   v4i __attribute__((ext_vector_type(4)));

#define NSAMP    32768
#define FEAT     576
#define TILE     16          // samples per workgroup tile (= WMMA M)
#define XS       580         // padded LDS row stride (580%64=4 -> conflict-free A-frag column reads)
#define NWAVES   8
#define NTHREADS 256
#define STJ      68          // t-tile stride between j components
#define STZ      352         // t-tile stride between z rows (>= 5*STJ, 16B aligned)

// irreps: 64x0e + 64x1o + 64x2e
__device__ static constexpr int IRR_OFF[3] = {0, 64, 256};
__device__ static constexpr int IRR_DIM[3] = {1, 3, 5};
// 9 (irrep, component) pairs for the linears
__device__ static constexpr int COMP_IRR[9] = {0,1,1,1,2,2,2,2,2};
__device__ static constexpr int COMP_IC[9]  = {0,0,1,2,0,1,2,3,4};

__device__ __forceinline__ v8f wmma_f32_k4(v2f a, v2f b, v8f c) {
  // D = A(16x4) * B(4x16) + C, full fp32 (ISA 15.10 op 93)
  return __builtin_amdgcn_wmma_f32_16x16x4_f32(false, a, false, b, (short)0, c,
                                               false, false);
}

// ------------------------- async global -> LDS tile fill --------------------
typedef __attribute__((address_space(1))) v4i* gv4p;  // global int4*
typedef __attribute__((address_space(3))) v4i* lv4p;  // LDS int4*

__device__ __forceinline__ void load_tile(const float* __restrict__ g, int z0,
                                          float* __restrict__ s, int tid) {
#if __has_builtin(__builtin_amdgcn_global_load_async_to_lds_b128)
  for (int q = tid; q < TILE * 144; q += NTHREADS) {
    int row = q / 144, c4 = (q % 144) * 4;
    gv4p src = (gv4p)(g + (size_t)(z0 + row) * FEAT + c4);
    lv4p dst = (lv4p)(s + row * XS + c4);
    __builtin_amdgcn_global_load_async_to_lds_b128(src, dst, 0, 0);
  }
#if __has_builtin(__builtin_amdgcn_s_wait_asynccnt)
  __builtin_amdgcn_s_wait_asynccnt(0);
#else
  asm volatile("s_wait_asynccnt 0" ::: "memory");
#endif
#else
  for (int q = tid; q < TILE * 144; q += NTHREADS) {
    int row = q / 144, c4 = (q % 144) * 4;
    float4 v = *(const float4*)(g + (size_t)(z0 + row) * FEAT + c4);
    *(float4*)(&s[row * XS + c4]) = v;
  }
#endif
}

// out[z, v] = sum_u arow[u*D] * wcol[u]   (wcol = transposed weight column,
// contiguous in K -> one b64 load per fragment; all offsets are literals)
template <int D>
__device__ __forceinline__ v8f linear_tile(const float* __restrict__ arow,
                                           const float* __restrict__ wcol,
                                           int kk) {
  v8f acc0 = {0.f,0.f,0.f,0.f,0.f,0.f,0.f,0.f};
  v8f acc1 = {0.f,0.f,0.f,0.f,0.f,0.f,0.f,0.f};
  const float* aK = arow + kk * D;   // lane-half K offset folded into base
  const float* wK = wcol + kk;
#pragma unroll
  for (int ku = 0; ku < 64; ku += 8) {
    v2f a0, a1;
    a0.x = aK[(ku + 0) * D];
    a0.y = aK[(ku + 1) * D];
    v2f b0 = *(const v2f*)(wK + ku);
    acc0 = wmma_f32_k4(a0, b0, acc0);
    a1.x = aK[(ku + 4) * D];
    a1.y = aK[(ku + 5) * D];
    v2f b1 = *(const v2f*)(wK + ku + 4);
    acc1 = wmma_f32_k4(a1, b1, acc1);
  }
  return acc0 + acc1;
}

__device__ __forceinline__ v8f linear_tile_d(int irr,
                                             const float* __restrict__ arow,
                                             const float* __restrict__ wcol,
                                             int kk) {
  if (irr == 0) return linear_tile<1>(arow, wcol, kk);
  if (irr == 1) return linear_tile<3>(arow, wcol, kk);
  return linear_tile<5>(arow, wcol, kk);
}

// ------------------ Kernel 0: transpose linear weights ----------------------
// wt[m][v*64+u] = w[m][u*64+v];  m = sel*3 + irr, sel: 0=lin, 1=s1, 2=s1a
__global__ void k_transpose_w(const float* __restrict__ lin_w,
                              const float* __restrict__ s1_w,
                              const float* __restrict__ s1a_w,
                              float* __restrict__ wt) {
  int m = blockIdx.x;
  const float* src = (m < 3) ? lin_w + m * 4096
                             : (m < 6) ? s1_w + (m - 3) * 4096
                                       : s1a_w + (m - 6) * 4096;
  float* dst = wt + m * 4096;
  for (int q = threadIdx.x; q < 4096; q += NTHREADS) {
    int u = q >> 6, v = q & 63;
    dst[v * 64 + u] = src[u * 64 + v];
  }
}

// --------------------------- Kernel 1: proj + silu --------------------------
__global__ __launch_bounds__(NTHREADS) void k_lin_silu(
    const float* __restrict__ x, const float* __restrict__ wt,
    float* __restrict__ pa) {
  __shared__ float sx[TILE * XS];
  const int tid = threadIdx.x, z0 = blockIdx.x * TILE;
  load_tile(x, z0, sx, tid);
  __syncthreads();
  const int wave = tid >> 5, lane = tid & 31, l16 = lane & 15, hf = lane >> 4,
            kk = hf * 2;
  for (int job = wave; job < 36; job += NWAVES) {   // wave-uniform: EXEC all-1s
    int comp = job >> 2, v0 = (job & 3) * 16;
    int irr = COMP_IRR[comp], ic = COMP_IC[comp];
    int off = IRR_OFF[irr], d = IRR_DIM[irr];
    const float* arow = sx + l16 * XS + off + ic;
    const float* wcol = wt + irr * 4096 + (v0 + l16) * 64;
    v8f acc = linear_tile_d(irr, arow, wcol, kk);
#pragma unroll
    for (int r = 0; r < 8; ++r) {
      int z = r + 8 * hf;
      float v = acc[r] * 0.125f;                    // / sqrt(64)
      v = v / (1.0f + __expf(-v));                  // silu
      pa[(size_t)(z0 + z) * FEAT + off + (v0 + l16) * d + ic] = v;
    }
  }
}

// ------------------- Kernel 2: h & y linears + BN stats ---------------------
__global__ __launch_bounds__(NTHREADS) void k_split_lin(
    const float* pa_in, const float* __restrict__ wt,
    const float* __restrict__ s1_b, const float* __restrict__ s1a_b,
    float* __restrict__ h_out, float* y_out, float* __restrict__ stats) {
  __shared__ float sp[TILE * XS];
  __shared__ float lstat[256];   // [0,192): sum y^2 per (irrep,ch); [192,256): sum y (l0)
  const int tid = threadIdx.x, z0 = blockIdx.x * TILE;
  load_tile(pa_in, z0, sp, tid);
  lstat[tid] = 0.f;
  __syncthreads();
  const int wave = tid >> 5, lane = tid & 31, l16 = lane & 15, hf = lane >> 4,
            kk = hf * 2;
  for (int job = wave; job < 72; job += NWAVES) {
    const bool isH = job < 36;
    int jj = isH ? job : job - 36;
    int comp = jj >> 2, v0 = (jj & 3) * 16;
    int irr = COMP_IRR[comp], ic = COMP_IC[comp];
    int off = IRR_OFF[irr], d = IRR_DIM[irr];
    const float* arow = sp + l16 * XS + off + ic;
    const float* wcol = wt + (isH ? 3 : 6) * 4096 + irr * 4096 + (v0 + l16) * 64;
    v8f acc = linear_tile_d(irr, arow, wcol, kk);
    float bias = (irr == 0) ? (isH ? s1_b : s1a_b)[v0 + l16] : 0.f;
    float* dst = isH ? h_out : y_out;   // y_out aliases pa_in tile (read done)
    float ssq = 0.f, ssum = 0.f;
#pragma unroll
    for (int r = 0; r < 8; ++r) {
      int z = r + 8 * hf;
      float v = acc[r] * 0.125f + bias;
      dst[(size_t)(z0 + z) * FEAT + off + (v0 + l16) * d + ic] = v;
      ssq += v * v;
      ssum += v;
    }
    if (!isH) {                                   // ds_add_f32 LDS atomics
      atomicAdd(&lstat[irr * 64 + v0 + l16], ssq);
      if (irr == 0) atomicAdd(&lstat[192 + v0 + l16], ssum);
    }
  }
  __syncthreads();
  atomicAdd(&stats[tid], lstat[tid]);             // one global atomic / channel
}

// ----------------------- Kernel 3: finalize BN params -----------------------
__global__ void k_bn_finalize(const float* __restrict__ stats,
                              const float* __restrict__ bn_w,
                              float* __restrict__ scale,
                              float* __restrict__ meanv) {
  int t = threadIdx.x;
  const float invN = 1.0f / (float)NSAMP;
  if (t < 64) {                                   // l=0: center, var
    float m = stats[192 + t] * invN;
    meanv[t] = m;
    float var = stats[t] * invN - m * m;
    scale[t] = bn_w[t] * rsqrtf(var + 1e-5f);
  } else if (t < 192) {                           // l>0: E[y^2] over (batch, dim)
    int irr = t >> 6;
    float dinv = (irr == 1) ? (1.f / 3.f) : (1.f / 5.f);
    float fn = stats[t] * invN * dinv;
    scale[t] = bn_w[t] * rsqrtf(fn + 1e-5f);
  }
}

// ---------------------- Kernel 4: BN+norm_act + tensor product --------------
struct CGE { int i, j, k; float c; };
// Wigner-3j sparse tables (e3nn real-SH convention, Frobenius-normalized;
// generated offline). l=1 basis (y,z,x); l=2 basis (xy,yz,z2,zx,x2-y2).
#define C3  0.57735027f
#define C5  0.44721360f
#define E6  0.40824829f
#define A1v 0.31622777f
#define B1v 0.36514837f
#define Bh  0.18257419f
#define P2  0.23904572f
#define Ph  0.11952286f
#define Q2  0.20701967f
__device__ static const CGE CG000[1]  = {{0,0,0,1.0f}};
__device__ static const CGE CG011[3]  = {{0,0,0,C3},{0,1,1,C3},{0,2,2,C3}};
__device__ static const CGE CG101[3]  = {{0,0,0,C3},{1,0,1,C3},{2,0,2,C3}};
__device__ static const CGE CG110[3]  = {{0,0,0,C3},{1,1,0,C3},{2,2,0,C3}};
__device__ static const CGE CG022[5]  = {{0,0,0,C5},{0,1,1,C5},{0,2,2,C5},{0,3,3,C5},{0,4,4,C5}};
__device__ static const CGE CG202[5]  = {{0,0,0,C5},{1,0,1,C5},{2,0,2,C5},{3,0,3,C5},{4,0,4,C5}};
__device__ static const CGE CG220[5]  = {{0,0,0,C5},{1,1,0,C5},{2,2,0,C5},{3,3,0,C5},{4,4,0,C5}};
__device__ static const CGE CG111[6]  = {{0,1,2,E6},{1,2,0,E6},{2,0,1,E6},
                                         {0,2,1,-E6},{2,1,0,-E6},{1,0,2,-E6}};
__device__ static const CGE CG112[11] = {{2,0,0,A1v},{0,2,0,A1v},{0,1,1,A1v},{1,0,1,A1v},
                                         {1,2,3,A1v},{2,1,3,A1v},{1,1,2,B1v},{0,0,2,-Bh},
                                         {2,2,2,-Bh},{2,2,4,A1v},{0,0,4,-A1v}};
__device__ static const CGE CG121[11] = {{2,0,0,A1v},{0,0,2,A1v},{0,1,1,A1v},{1,1,0,A1v},
                                         {1,3,2,A1v},{2,3,1,A1v},{1,2,1,B1v},{0,2,0,-Bh},
                                         {2,2,2,-Bh},{2,4,2,A1v},{0,4,0,-A1v}};
__device__ static const CGE CG211[11] = {{0,2,0,A1v},{0,0,2,A1v},{1,0,1,A1v},{1,1,0,A1v},
                                         {3,1,2,A1v},{3,2,1,A1v},{2,1,1,B1v},{2,0,0,-Bh},
                                         {2,2,2,-Bh},{4,2,2,A1v},{4,0,0,-A1v}};
__device__ static const CGE CG222[25] = {
  {0,0,2,-P2},{0,2,0,-P2},{2,0,0,-P2},
  {1,1,2,Ph},{1,2,1,Ph},{2,1,1,Ph},
  {3,3,2,Ph},{3,2,3,Ph},{2,3,3,Ph},
  {2,2,2,P2},
  {2,4,4,-P2},{4,2,4,-P2},{4,4,2,-P2},
  {1,1,4,-Q2},{1,4,1,-Q2},{4,1,1,-Q2},
  {3,3,4,Q2},{3,4,3,Q2},{4,3,3,Q2},
  {0,1,3,Q2},{0,3,1,Q2},{1,0,3,Q2},{1,3,0,Q2},{3,0,1,Q2},{3,1,0,Q2}};

// t[z,u,j] = alpha * sum_v tp_w[p][u][v] * ha[z, off2 + v*D2 + j]
// B[K=v][N=u] = tp_w[u][v]: K-contiguous in memory -> b64 fragment loads.
template <int D2>
__device__ __forceinline__ void tp_mix(const float* __restrict__ sa,
                                       float* __restrict__ st,
                                       const float* __restrict__ wp, float alpha,
                                       int off2, int wave, int l16, int hf,
                                       int kk) {
  for (int job = wave; job < D2 * 4; job += NWAVES) {
    int j = job >> 2, u0 = (job & 3) * 16;
    v8f acc0 = {0.f,0.f,0.f,0.f,0.f,0.f,0.f,0.f};
    v8f acc1 = {0.f,0.f,0.f,0.f,0.f,0.f,0.f,0.f};
    const float* aK = sa + l16 * XS + off2 + j + kk * D2;
    const float* wK = wp + (u0 + l16) * 64 + kk;
#pragma unroll
    for (int ku = 0; ku < 64; ku += 8) {
      v2f a0, a1;
      a0.x = aK[(ku + 0) * D2];
      a0.y = aK[(ku + 1) * D2];
      v2f b0 = *(const v2f*)(wK + ku);
      acc0 = wmma_f32_k4(a0, b0, acc0);
      a1.x = aK[(ku + 4) * D2];
      a1.y = aK[(ku + 5) * D2];
      v2f b1 = *(const v2f*)(wK + ku + 4);
      acc1 = wmma_f32_k4(a1, b1, acc1);
    }
    v8f acc = acc0 + acc1;
#pragma unroll
    for (int r = 0; r < 8; ++r)
      st[(r + 8 * hf) * STZ + j * STJ + u0 + l16] = alpha * acc[r];
  }
}

template <int NE>
__device__ __forceinline__ void tp_acc(const CGE (&L)[NE],
                                       const float (&hreg)[4][9],
                                       float (&oacc)[4][9],
                                       const float* __restrict__ st, int base,
                                       int hoff, int ooff) {
#pragma unroll
  for (int q = 0; q < 4; ++q) {
    int idx = base + q, z = idx >> 6, u = idx & 63;
    const float* tp = st + z * STZ + u;
#pragma unroll
    for (int e = 0; e < NE; ++e)
      oacc[q][ooff + L[e].k] += L[e].c * hreg[q][hoff + L[e].i] * tp[L[e].j * STJ];
  }
}

__global__ __launch_bounds__(NTHREADS) void k_tensor_product(
    const float* __restrict__ h_in, const float* __restrict__ y_in,
    const float* __restrict__ tp_w, const float* __restrict__ scale,
    const float* __restrict__ meanv, const float* __restrict__ bn_b,
    float* __restrict__ out) {
  __shared__ float sa[TILE * XS];   // ha tile (post BN + norm_act)
  __shared__ float st[TILE * STZ];  // per-path mixed tile t
  const int tid = threadIdx.x, z0 = blockIdx.x * TILE;
  const int wave = tid >> 5, lane = tid & 31, l16 = lane & 15, hf = lane >> 4,
            kk = hf * 2;
  const int base = tid * 4;         // 4 (z,u) pairs per thread
  float hreg[4][9], oacc[4][9];
#pragma unroll
  for (int q = 0; q < 4; ++q) {
    int idx = base + q, z = idx >> 6, u = idx & 63;
    const float* hrow = h_in + (size_t)(z0 + z) * FEAT;
    const float* yrow = y_in + (size_t)(z0 + z) * FEAT;
    // l=0: center + scale + beta, then norm_act
    hreg[q][0] = hrow[u];
    float f0 = (yrow[u] - meanv[u]) * scale[u] + bn_b[u];
    float n0 = sqrtf(f0 * f0 + 1e-16f);
    sa[z * XS + u] = f0 * (1.f / (1.f + __expf(-n0))) / n0;
    // l=1
    float f[5], nn = 1e-16f, s1 = scale[64 + u];
#pragma unroll
    for (int i = 0; i < 3; ++i) {
      hreg[q][1 + i] = hrow[64 + u * 3 + i];
      f[i] = yrow[64 + u * 3 + i] * s1;
      nn += f[i] * f[i];
    }
    float n1 = sqrtf(nn), g1 = (1.f / (1.f + __expf(-n1))) / n1;
#pragma unroll
    for (int i = 0; i < 3; ++i) sa[z * XS + 64 + u * 3 + i] = f[i] * g1;
    // l=2
    nn = 1e-16f;
    float s2 = scale[128 + u];
#pragma unroll
    for (int i = 0; i < 5; ++i) {
      hreg[q][4 + i] = hrow[256 + u * 5 + i];
      f[i] = yrow[256 + u * 5 + i] * s2;
      nn += f[i] * f[i];
    }
    float n2 = sqrtf(nn), g2 = (1.f / (1.f + __expf(-n2))) / n2;
#pragma unroll
    for (int i = 0; i < 5; ++i) sa[z * XS + 256 + u * 5 + i] = f[i] * g2;
#pragma unroll
    for (int k = 0; k < 9; ++k) oacc[q][k] = 0.f;
  }
  __syncthreads();

  // path alphas: sqrt(d_out / (64 * npaths(out)))
  const float a0 = 0.07216878f, a1 = 0.10825318f, a2 = 0.13975425f;
  const int HOFF[3] = {0, 1, 4};
#define DO_PATH(P, I1, I2, IO, LIST, ALPHA)                                    \
  tp_mix<IRR_DIM[I2]>(sa, st, tp_w + (P) * 4096, (ALPHA), IRR_OFF[I2], wave,   \
                      l16, hf, kk);                                            \
  __syncthreads();                                                             \
  tp_acc(LIST, hreg, oacc, st, base, HOFF[I1], HOFF[IO]);                      \
  __syncthreads();

  DO_PATH(0, 0, 0, 0, CG000, a0)
  DO_PATH(1, 0, 1, 1, CG011, a1)
  DO_PATH(2, 0, 2, 2, CG022, a2)
  DO_PATH(3, 1, 0, 1, CG101, a1)
  DO_PATH(4, 1, 1, 0, CG110, a0)
  DO_PATH(5, 1, 1, 2, CG112, a2)
  DO_PATH(6, 1, 2, 1, CG121, a1)
  DO_PATH(7, 2, 0, 2, CG202, a2)
  DO_PATH(8, 2, 1, 1, CG211, a1)
  DO_PATH(9, 2, 2, 0, CG220, a0)
  DO_PATH(10, 2, 2, 2, CG222, a2)
#undef DO_PATH

#pragma unroll
  for (int q = 0; q < 4; ++q) {
    int idx = base + q, z = idx >> 6, u = idx & 63;
    float* orow = out + (size_t)(z0 + z) * FEAT;
    orow[u] = oacc[q][0];
#pragma unroll
    for (int i = 0; i < 3; ++i) orow[64 + u * 3 + i] = oacc[q][1 + i];
#pragma unroll
    for (int i = 0; i < 5; ++i) orow[256 + u * 5 + i] = oacc[q][4 + i];
  }
}

// ---------------------------------------------------------------------------
extern "C" void kernel_launch(void* const* d_in, const int* in_sizes, int n_in,
                              void* d_out, int out_size, void* d_ws,
                              size_t ws_size, hipStream_t stream) {
  const float* x     = (const float*)d_in[0];
  const float* lin_w = (const float*)d_in[1];
  const float* s1_w  = (const float*)d_in[2];
  const float* s1_b  = (const float*)d_in[3];
  const float* s1a_w = (const float*)d_in[4];
  const float* s1a_b = (const float*)d_in[5];
  const float* bn_w  = (const float*)d_in[6];
  const float* bn_b  = (const float*)d_in[7];
  const float* tp_w  = (const float*)d_in[8];
  float* out = (float*)d_out;
  float* ws  = (float*)d_ws;

  const size_t NF = (size_t)NSAMP * FEAT;
  float* pa    = ws;            // proj_a; reused as y after k_split_lin
  float* hbuf  = ws + NF;
  float* stats = ws + 2 * NF;   // 256 floats (sum y^2 per channel, sum y l0)
  float* scale = stats + 256;   // 192 floats
  float* meanv = scale + 192;   // 64 floats
  float* wt    = meanv + 64;    // 9 transposed 64x64 weight matrices

  (void)hipMemsetAsync(stats, 0, 256 * sizeof(float), stream);

  dim3 grid(NSAMP / TILE), blk(NTHREADS);
  k_transpose_w<<<9, blk, 0, stream>>>(lin_w, s1_w, s1a_w, wt);
  k_lin_silu<<<grid, blk, 0, stream>>>(x, wt, pa);
  k_split_lin<<<grid, blk, 0, stream>>>(pa, wt, s1_b, s1a_b, hbuf, pa, stats);
  k_bn_finalize<<<1, 256, 0, stream>>>(stats, bn_w, scale, meanv);
  k_tensor_product<<<grid, blk, 0, stream>>>(hbuf, pa, tp_w, scale, meanv,
                                             bn_b, out);
}